// EMACalculator_21784074125426
// MI455X (gfx1250) — compile-verified
//
#include <hip/hip_runtime.h>
#include <hip/hip_bf16.h>

// EMA gated scan: x (32,4096,256) f32, mask (32,4096) i32 -> out (32,4096,256) f32
// 3-phase chunked linear-recurrence scan; phase 3 uses V_WMMA_F32_16X16X4_F32
// to evaluate each 16-step window as a lower-triangular 16x16 matmul.

#define EB     32      // batch
#define ET     4096    // time
#define ED     256     // feature
#define ALPHA  0.2f
#define ECH    64              // chunk length (timesteps)
#define ENC    (ET / ECH)      // 64 chunks
#define ENW    (ECH / 16)      // 4 windows of 16 per chunk

typedef __attribute__((ext_vector_type(2))) float v2f;
typedef __attribute__((ext_vector_type(8))) float v8f;

__device__ __forceinline__ void coeffs(int tg, int v, float& a, float& al) {
    // e_0 = x_0 regardless of mask -> a=0, al=1 at global t==0
    if (tg == 0) { a = 0.0f; al = 1.0f; }
    else { a = v ? (1.0f - ALPHA) : 1.0f; al = v ? ALPHA : 0.0f; }
}

// ---------------- Phase 1: per-chunk aggregates -----------------------------
// A_c = prod a_t over chunk (scalar per (b,c)); B_c[d] = chunk applied to zero carry.
__global__ __launch_bounds__(256) void ema_p1_agg(const float* __restrict__ x,
                                                  const int* __restrict__ mask,
                                                  float* __restrict__ aggA,
                                                  float* __restrict__ aggB) {
    const int c = blockIdx.x, b = blockIdx.y, d = threadIdx.x;
    __shared__ float sA[ECH], sAl[ECH];
    const int t0 = c * ECH;
    if (d < ECH) {
        float a, al;
        coeffs(t0 + d, mask[b * ET + t0 + d], a, al);
        sA[d] = a; sAl[d] = al;
    }
    __syncthreads();
    const float* xp = x + ((size_t)(b * ET + t0)) * ED + d;
    float e = 0.0f, aa = 1.0f;
#pragma unroll 8
    for (int t = 0; t < ECH; ++t) {
        if ((t & 7) == 0) __builtin_prefetch(xp + (t + 24) * ED, 0, 0);
        float xv = xp[t * ED];
        e = fmaf(sA[t], e, sAl[t] * xv);
        aa *= sA[t];
    }
    aggB[((size_t)b * ENC + c) * ED + d] = e;
    if (d == 0) aggA[b * ENC + c] = aa;
}

// ---------------- Phase 2: sequential chunk-carry combine -------------------
// carry[c][d] = EMA value just before chunk c starts. Tiny: 32 blocks x 256 thr.
__global__ __launch_bounds__(256) void ema_p2_carry(const float* __restrict__ aggA,
                                                    const float* __restrict__ aggB,
                                                    float* __restrict__ carry) {
    const int b = blockIdx.x, d = threadIdx.x;
    float cr = 0.0f;
    carry[((size_t)b * ENC + 0) * ED + d] = 0.0f;
    for (int c = 1; c < ENC; ++c) {
        float A  = aggA[b * ENC + c - 1];
        float Bv = aggB[((size_t)b * ENC + c - 1) * ED + d];
        cr = fmaf(A, cr, Bv);
        carry[((size_t)b * ENC + c) * ED + d] = cr;
    }
}

// ---------------- Phase 3: WMMA windowed scan -------------------------------
// Per (b, chunk) workgroup: 8 waves, each owns two 16-wide d-tiles.
// Window math: E(16x16) = M(16x16) @ X(16x16) + p (x) carry, carry' = E[15].
__global__ __launch_bounds__(256) void ema_p3_wmma(const float* __restrict__ x,
                                                   const int* __restrict__ mask,
                                                   const float* __restrict__ carry,
                                                   float* __restrict__ out) {
    const int c = blockIdx.x, b = blockIdx.y;
    const int tid  = threadIdx.x;
    const int lane = tid & 31;
    const int wave = tid >> 5;       // 0..7
    const int lh   = lane >> 4;      // lane half (0/1)
    const int ln   = lane & 15;      // N / row index within half

    __shared__ float sA[16], sAl[16];
    __shared__ float sM[16 * 16];    // row-major coefficient matrix
    __shared__ float sP[16];         // carry coefficients p_t

    const int t0c = c * ECH;
    const int db0 = (wave * 2 + 0) * 16;
    const int db1 = (wave * 2 + 1) * 16;

    float cr0 = carry[((size_t)b * ENC + c) * ED + db0 + ln];
    float cr1 = carry[((size_t)b * ENC + c) * ED + db1 + ln];

    for (int win = 0; win < ENW; ++win) {
        const int t0 = t0c + win * 16;

        // Stage per-step coefficients (16 threads).
        if (tid < 16) {
            float a, al;
            coeffs(t0 + tid, mask[b * ET + t0 + tid], a, al);
            sA[tid] = a; sAl[tid] = al;
        }
        __syncthreads();

        // Build M row t: M[t][s] = al_s * prod_{u=s+1..t} a_u ; p_t = prod_{u<=t} a_u.
        if (tid < 16) {
            const int t = tid;
            float q = 1.0f;
            for (int s = t; s >= 0; --s) { sM[t * 16 + s] = sAl[s] * q; q *= sA[s]; }
            for (int s = t + 1; s < 16; ++s) sM[t * 16 + s] = 0.0f;
            sP[t] = q;
        }
        __syncthreads();

        // A fragments (shared by both d-tiles). A layout (16x4 f32 slice):
        // vgpr0: col 4k+2*lh, vgpr1: col 4k+2*lh+1, row = ln  -> one aligned b64 read.
        v2f afrag[4];
#pragma unroll
        for (int k = 0; k < 4; ++k)
            afrag[k] = *reinterpret_cast<const v2f*>(&sM[ln * 16 + 4 * k + 2 * lh]);

        float pv[8];
#pragma unroll
        for (int v = 0; v < 8; ++v) pv[v] = sP[8 * lh + v];

        // B per-lane base: row = t0 + 2*lh (+ uniform immediate offsets), col = db+ln.
        const size_t rowbase = (size_t)(b * ET + t0 + 2 * lh) * ED;
        const float* bp0 = x + rowbase + db0 + ln;
        const float* bp1 = x + rowbase + db1 + ln;
        __builtin_prefetch(bp0 + 16 * ED, 0, 0);   // next window

        // ---- tile 0 ----
        {
            v8f acc = {};
#pragma unroll
            for (int k = 0; k < 4; ++k) {
                v2f bf;
                bf.x = bp0[(4 * k + 0) * ED];
                bf.y = bp0[(4 * k + 1) * ED];
                acc = __builtin_amdgcn_wmma_f32_16x16x4_f32(
                          false, afrag[k], false, bf, (short)0, acc, false, false);
            }
#pragma unroll
            for (int v = 0; v < 8; ++v) acc[v] = fmaf(pv[v], cr0, acc[v]);
            float* op = out + (size_t)(b * ET + t0 + 8 * lh) * ED + db0 + ln;
#pragma unroll
            for (int v = 0; v < 8; ++v) op[v * ED] = acc[v];
            // row 15 lives in vgpr7 of lanes 16..31; broadcast col ln to both halves
            cr0 = __shfl(acc[7], 16 + ln, 32);
        }
        // ---- tile 1 ----
        {
            v8f acc = {};
#pragma unroll
            for (int k = 0; k < 4; ++k) {
                v2f bf;
                bf.x = bp1[(4 * k + 0) * ED];
                bf.y = bp1[(4 * k + 1) * ED];
                acc = __builtin_amdgcn_wmma_f32_16x16x4_f32(
                          false, afrag[k], false, bf, (short)0, acc, false, false);
            }
#pragma unroll
            for (int v = 0; v < 8; ++v) acc[v] = fmaf(pv[v], cr1, acc[v]);
            float* op = out + (size_t)(b * ET + t0 + 8 * lh) * ED + db1 + ln;
#pragma unroll
            for (int v = 0; v < 8; ++v) op[v * ED] = acc[v];
            cr1 = __shfl(acc[7], 16 + ln, 32);
        }
        __syncthreads();   // protect sM/sP before next window's rebuild
    }
}

// ---------------------------------------------------------------------------
extern "C" void kernel_launch(void* const* d_in, const int* in_sizes, int n_in,
                              void* d_out, int out_size, void* d_ws, size_t ws_size,
                              hipStream_t stream) {
    const float* x    = (const float*)d_in[0];
    const int*   mask = (const int*)d_in[1];
    float*       out  = (float*)d_out;

    // Workspace: aggA (B*NC, padded to 4096 floats) | aggB (B*NC*D) | carry (B*NC*D)
    float* ws    = (float*)d_ws;
    float* aggA  = ws;
    float* aggB  = ws + 4096;
    float* carry = aggB + (size_t)EB * ENC * ED;

    dim3 gridBC(ENC, EB);
    ema_p1_agg  <<<gridBC,   256, 0, stream>>>(x, mask, aggA, aggB);
    ema_p2_carry<<<dim3(EB), 256, 0, stream>>>(aggA, aggB, carry);
    ema_p3_wmma <<<gridBC,   256, 0, stream>>>(x, mask, carry, out);
    (void)in_sizes; (void)n_in; (void)out_size; (void)ws_size;
}